// QuantumAttentionMechanism_79551384257181
// MI455X (gfx1250) — compile-verified
//
#include <hip/hip_runtime.h>
#include <hip/hip_bf16.h>
#include <math.h>

typedef float v2f __attribute__((ext_vector_type(2)));
typedef float v8f __attribute__((ext_vector_type(8)));
typedef unsigned int u32x4 __attribute__((ext_vector_type(4)));
typedef int i32x4 __attribute__((ext_vector_type(4)));
typedef int i32x8 __attribute__((ext_vector_type(8)));

static constexpr int kB   = 2;
static constexpr int kS   = 2048;
static constexpr int kDIM = 1024;
static constexpr int kH   = 16;
static constexpr int kD   = 64;
static constexpr int kBS  = kB * kS;      // 4096 token rows
static constexpr int kG   = kDIM / 2;     // 512 gate hidden

static __device__ __forceinline__ v8f wmma_f32(v2f a, v2f b, v8f c) {
    return __builtin_amdgcn_wmma_f32_16x16x4_f32(
        /*neg_a=*/false, a, /*neg_b=*/false, b,
        /*c_mod=*/(short)0, c, /*reuse_a=*/false, /*reuse_b=*/false);
}

// ---------------------------------------------------------------------------
// TDM: issue a 2D tile load Global -> LDS via tensor descriptor (D#).
// Group 0: [1:0]=count(1), [63:32]=lds_addr, [120:64]=global_addr, [127:126]=type(2)
// Group 1: [17:16]=data_size(2=4B), [20]=pad_enable, [24:22]=pad_interval,
//          [31:25]=pad_amount, [79:48]=tensor_dim0, [111:80]=tensor_dim1,
//          [127:112]=tile_dim0, [143:128]=tile_dim1, [207:160]=tensor_dim0_stride
// pad_interval code: 4 -> every 32 DWORDs, 5 -> every 64 DWORDs; amount 0 -> 1 DWORD
// ---------------------------------------------------------------------------
static __device__ __forceinline__ void tdm_load_2d(
    unsigned lds_addr, const void* gptr,
    unsigned tile_w, unsigned tile_h,
    unsigned tensor_w, unsigned tensor_h, unsigned row_stride,
    unsigned pad_interval, unsigned pad_amount)
{
    unsigned long long ga = (unsigned long long)(size_t)gptr;
    u32x4 g0;
    g0[0] = 1u;                                           // count=1, user desc
    g0[1] = lds_addr;                                     // LDS byte address
    g0[2] = (unsigned)ga;                                 // global addr [31:0]
    g0[3] = (unsigned)((ga >> 32) & 0x01FFFFFFull)        // global addr [56:32]
          | (2u << 30);                                   // type=2 ("image")
    i32x8 g1;
    g1[0] = (int)((2u << 16) | (1u << 20) |
                  (pad_interval << 22) | (pad_amount << 25));
    g1[1] = (int)((tensor_w & 0xFFFFu) << 16);                               // dim0 lo
    g1[2] = (int)(((tensor_w >> 16) & 0xFFFFu) | ((tensor_h & 0xFFFFu) << 16)); // dim0 hi | dim1 lo
    g1[3] = (int)(((tensor_h >> 16) & 0xFFFFu) | ((tile_w & 0xFFFFu) << 16));   // dim1 hi | tile_dim0
    g1[4] = (int)(tile_h & 0xFFFFu);                                         // tile_dim1 (tile_dim2=0)
    g1[5] = (int)row_stride;                                                 // dim0_stride [31:0]
    g1[6] = 0;                                                               // dim0_stride hi | dim1_stride lo
    g1[7] = 0;
    i32x4 gz4 = {0, 0, 0, 0};
    i32x8 gz8 = {0, 0, 0, 0, 0, 0, 0, 0};
    // 6-arg form (clang-23 / therock headers): (g0, g1, g2, g3, g4, cpol)
    __builtin_amdgcn_tensor_load_to_lds(g0, g1, gz4, gz4, gz8, 0);
}

static __device__ __forceinline__ unsigned lds_off(const void* p) {
    return (unsigned)(size_t)p;   // low 32 bits of generic LDS address = DS offset
}

// ---------------------------------------------------------------------------
// C[M,N] = act(A[M,K] @ W[K,N] + bias[N]); act: 0 = identity, 1 = tanh
// Block tile 64x64, 8 waves; TDM double-buffered K-chunks of 32.
// ---------------------------------------------------------------------------
__global__ __launch_bounds__(256)
void gemm_bias_act(const float* __restrict__ A, const float* __restrict__ W,
                   const float* __restrict__ bias, float* __restrict__ C,
                   int M, int N, int K, int act) {
    __shared__ float As[2][64][33];   // [buf][m][k], TDM pads 1 DWORD / 32
    __shared__ float Ws[2][32][65];   // [buf][k][n], TDM pads 1 DWORD / 64

    const int tid  = threadIdx.x;
    const int lane = tid & 31;
    const int wave = tid >> 5;
    const int wm   = wave & 3;          // 0..3
    const int wn   = wave >> 2;         // 0..1
    const int row0 = blockIdx.x * 64;
    const int col0 = blockIdx.y * 64;

    const int mf = lane & 15;           // fragment row / col within 16
    const int kf = (lane >> 4) * 2;     // fragment k base (0 or 2)

    // Prologue: DMA chunk 0 into buffer 0 (wave-level op, issued once by wave 0)
    if (wave == 0) {
        tdm_load_2d(lds_off(&As[0][0][0]), A + (size_t)row0 * K,
                    /*tile*/32, 64, /*tensor*/K, M, /*stride*/K, /*pad*/4, 0);
        tdm_load_2d(lds_off(&Ws[0][0][0]), W + col0,
                    /*tile*/64, 32, /*tensor*/N, K, /*stride*/N, /*pad*/5, 0);
    }

    v8f acc0 = {};
    v8f acc1 = {};

    for (int k0 = 0; k0 < K; k0 += 32) {
        const int cur = (k0 >> 5) & 1;
        const int nxt = cur ^ 1;
        if (wave == 0) {
            if (k0 + 32 < K) {   // prefetch next chunk via TDM, then wait current
                tdm_load_2d(lds_off(&As[nxt][0][0]),
                            A + (size_t)row0 * K + k0 + 32,
                            32, 64, K, M, K, 4, 0);
                tdm_load_2d(lds_off(&Ws[nxt][0][0]),
                            W + (size_t)(k0 + 32) * N + col0,
                            64, 32, N, K, N, 5, 0);
                __builtin_amdgcn_s_wait_tensorcnt(2);   // current chunk complete
            } else {
                __builtin_amdgcn_s_wait_tensorcnt(0);   // drain
            }
        }
        __syncthreads();

        #pragma unroll
        for (int ks = 0; ks < 32; ks += 4) {
            v2f a, b0, b1;
            a.x  = As[cur][wm * 16 + mf][ks + kf];
            a.y  = As[cur][wm * 16 + mf][ks + kf + 1];
            b0.x = Ws[cur][ks + kf][wn * 32 + mf];
            b0.y = Ws[cur][ks + kf + 1][wn * 32 + mf];
            b1.x = Ws[cur][ks + kf][wn * 32 + 16 + mf];
            b1.y = Ws[cur][ks + kf + 1][wn * 32 + 16 + mf];
            acc0 = wmma_f32(a, b0, acc0);
            acc1 = wmma_f32(a, b1, acc1);
        }
        __syncthreads();
    }

    // C/D layout: VGPR r -> row r (lanes 0-15) / r+8 (lanes 16-31), col = lane&15
    const int hf = lane >> 4;
    #pragma unroll
    for (int r = 0; r < 8; ++r) {
        int mrow = wm * 16 + r + hf * 8;
        int ncol = wn * 32 + mf;
        float v0 = acc0[r] + bias[col0 + ncol];
        float v1 = acc1[r] + bias[col0 + ncol + 16];
        if (act) { v0 = tanhf(v0); v1 = tanhf(v1); }
        C[(size_t)(row0 + mrow) * N + col0 + ncol]      = v0;
        C[(size_t)(row0 + mrow) * N + col0 + ncol + 16] = v1;
    }
}

// ---------------------------------------------------------------------------
// gate[row, h] = softmax_h(G1[row,:] @ Wg2[:,h] + bg2[h]); one wave per 16 rows
// ---------------------------------------------------------------------------
__global__ __launch_bounds__(32)
void gate_kernel(const float* __restrict__ G1, const float* __restrict__ Wg2,
                 const float* __restrict__ bg2, float* __restrict__ gate) {
    const int lane = threadIdx.x;
    const int row0 = blockIdx.x * 16;
    const int n  = lane & 15;
    const int kf = (lane >> 4) * 2;
    const int hf = lane >> 4;

    v8f acc = {};
    for (int k0 = 0; k0 < kG; k0 += 4) {
        v2f a, b;
        a.x = G1[(size_t)(row0 + n) * kG + k0 + kf];
        a.y = G1[(size_t)(row0 + n) * kG + k0 + kf + 1];
        b.x = Wg2[(size_t)(k0 + kf) * kH + n];
        b.y = Wg2[(size_t)(k0 + kf + 1) * kH + n];
        acc = wmma_f32(a, b, acc);
    }

    const float bn = bg2[n];
    #pragma unroll
    for (int r = 0; r < 8; ++r) {
        float v = acc[r] + bn;
        // softmax over the 16 columns of this row (within each 16-lane half)
        float mx = v;
        #pragma unroll
        for (int off = 8; off >= 1; off >>= 1) mx = fmaxf(mx, __shfl_xor(mx, off, 32));
        float e = expf(v - mx);
        float s = e;
        #pragma unroll
        for (int off = 8; off >= 1; off >>= 1) s += __shfl_xor(s, off, 32);
        gate[(size_t)(row0 + r + hf * 8) * kH + n] = e / s;
    }
}

// ---------------------------------------------------------------------------
// Flash attention with per-query gate scale applied to raw scores.
// grid = (S/128, H, B), block = 256 (8 waves x 16 query rows)
// ---------------------------------------------------------------------------
__global__ __launch_bounds__(256)
void attention_kernel(const float* __restrict__ Q, const float* __restrict__ Kb,
                      const float* __restrict__ Vb, const float* __restrict__ gate,
                      float* __restrict__ attn) {
    __shared__ float Ks[16][65];        // [key][dim]
    __shared__ float Vs[16][65];        // [key][dim]
    __shared__ float Ps[8][16][17];     // per-wave probs tile [row][key]

    const int tid  = threadIdx.x;
    const int lane = tid & 31;
    const int wave = tid >> 5;
    const int h  = blockIdx.y;
    const int bb = blockIdx.z;
    const int q0 = blockIdx.x * 128 + wave * 16;

    const int mf = lane & 15;
    const int hf = lane >> 4;
    const int kf = hf * 2;

    const size_t base = (size_t)bb * kS * kDIM + (size_t)h * kD;

    // Preload Q A-fragments for all 16 k-steps (full d=64)
    float qa[16][2];
    #pragma unroll
    for (int s = 0; s < 16; ++s) {
        size_t qi = base + (size_t)(q0 + mf) * kDIM + 4 * s + kf;
        qa[s][0] = Q[qi];
        qa[s][1] = Q[qi + 1];
    }
    // Gate (pre-softmax score scale) per accumulator row, 1/sqrt(64) folded in
    float gq[8];
    #pragma unroll
    for (int r = 0; r < 8; ++r)
        gq[r] = gate[((size_t)bb * kS + q0 + r + hf * 8) * kH + h] * 0.125f;

    v8f o0 = {}, o1 = {}, o2 = {}, o3 = {};
    float mrow[8], lrow[8];
    #pragma unroll
    for (int r = 0; r < 8; ++r) { mrow[r] = -1e30f; lrow[r] = 0.0f; }

    for (int j0 = 0; j0 < kS; j0 += 16) {
        // Cooperative stage of K and V 16x64 tiles
        #pragma unroll
        for (int i = tid; i < 16 * 64; i += 256) {
            int kr = i >> 6;
            int cc = i & 63;
            Ks[kr][cc] = Kb[base + (size_t)(j0 + kr) * kDIM + cc];
            Vs[kr][cc] = Vb[base + (size_t)(j0 + kr) * kDIM + cc];
        }
        if (j0 + 16 < kS) {   // hint next tile -> global_prefetch_b8
            __builtin_prefetch(Kb + base + (size_t)(j0 + 16 + (tid & 15)) * kDIM, 0, 1);
            __builtin_prefetch(Vb + base + (size_t)(j0 + 16 + (tid & 15)) * kDIM, 0, 1);
        }
        __syncthreads();

        // scores = Q (16x64) @ K^T (64x16): B[k][n] = Ks[n][k]
        v8f sc = {};
        #pragma unroll
        for (int s = 0; s < 16; ++s) {
            v2f a, b;
            a.x = qa[s][0];
            a.y = qa[s][1];
            b.x = Ks[mf][4 * s + kf];
            b.y = Ks[mf][4 * s + kf + 1];
            sc = wmma_f32(a, b, sc);
        }

        // Online softmax with gate scale (pre-softmax, per query row)
        #pragma unroll
        for (int r = 0; r < 8; ++r) {
            float sv = sc[r] * gq[r];
            float mx = sv;
            #pragma unroll
            for (int off = 8; off >= 1; off >>= 1) mx = fmaxf(mx, __shfl_xor(mx, off, 32));
            float mnew  = fmaxf(mrow[r], mx);
            float alpha = expf(mrow[r] - mnew);
            float p     = expf(sv - mnew);
            float psum  = p;
            #pragma unroll
            for (int off = 8; off >= 1; off >>= 1) psum += __shfl_xor(psum, off, 32);
            lrow[r] = lrow[r] * alpha + psum;
            mrow[r] = mnew;
            o0[r] *= alpha; o1[r] *= alpha; o2[r] *= alpha; o3[r] *= alpha;
            Ps[wave][r + hf * 8][mf] = p;   // re-layout probs via LDS
        }

        // O += P (16x16) @ V (16x64): 4 k-steps x 4 column tiles
        #pragma unroll
        for (int s = 0; s < 4; ++s) {
            v2f a, b0, b1, b2, b3;
            a.x  = Ps[wave][mf][4 * s + kf];
            a.y  = Ps[wave][mf][4 * s + kf + 1];
            b0.x = Vs[4 * s + kf][mf];           b0.y = Vs[4 * s + kf + 1][mf];
            b1.x = Vs[4 * s + kf][16 + mf];      b1.y = Vs[4 * s + kf + 1][16 + mf];
            b2.x = Vs[4 * s + kf][32 + mf];      b2.y = Vs[4 * s + kf + 1][32 + mf];
            b3.x = Vs[4 * s + kf][48 + mf];      b3.y = Vs[4 * s + kf + 1][48 + mf];
            o0 = wmma_f32(a, b0, o0);
            o1 = wmma_f32(a, b1, o1);
            o2 = wmma_f32(a, b2, o2);
            o3 = wmma_f32(a, b3, o3);
        }
        __syncthreads();
    }

    // Normalize and write attended (b, s, h, d) == (BS, DIM) row-major
    #pragma unroll
    for (int r = 0; r < 8; ++r) {
        float inv = 1.0f / lrow[r];
        size_t ob = base + (size_t)(q0 + r + hf * 8) * kDIM;
        attn[ob + mf]      = o0[r] * inv;
        attn[ob + 16 + mf] = o1[r] * inv;
        attn[ob + 32 + mf] = o2[r] * inv;
        attn[ob + 48 + mf] = o3[r] * inv;
    }
}

// ---------------------------------------------------------------------------
extern "C" void kernel_launch(void* const* d_in, const int* in_sizes, int n_in,
                              void* d_out, int out_size, void* d_ws, size_t ws_size,
                              hipStream_t stream) {
    const float* x   = (const float*)d_in[0];
    const float* Wq  = (const float*)d_in[1];
    const float* bq  = (const float*)d_in[2];
    const float* Wk  = (const float*)d_in[3];
    const float* bk  = (const float*)d_in[4];
    const float* Wv  = (const float*)d_in[5];
    const float* bv  = (const float*)d_in[6];
    const float* Wg1 = (const float*)d_in[7];
    const float* bg1 = (const float*)d_in[8];
    const float* Wg2 = (const float*)d_in[9];
    const float* bg2 = (const float*)d_in[10];
    const float* Wo  = (const float*)d_in[11];
    const float* bo  = (const float*)d_in[12];
    float* out = (float*)d_out;

    float* Qb   = (float*)d_ws;
    float* Kbuf = Qb   + (size_t)kBS * kDIM;
    float* Vbuf = Kbuf + (size_t)kBS * kDIM;
    float* G1   = Vbuf + (size_t)kBS * kDIM;
    float* gate = G1   + (size_t)kBS * kG;
    float* attn = gate + (size_t)kBS * kH;

    dim3 blk(256);
    // QKV + gate-hidden projections
    gemm_bias_act<<<dim3(kBS / 64, kDIM / 64), blk, 0, stream>>>(x, Wq, bq, Qb,   kBS, kDIM, kDIM, 0);
    gemm_bias_act<<<dim3(kBS / 64, kDIM / 64), blk, 0, stream>>>(x, Wk, bk, Kbuf, kBS, kDIM, kDIM, 0);
    gemm_bias_act<<<dim3(kBS / 64, kDIM / 64), blk, 0, stream>>>(x, Wv, bv, Vbuf, kBS, kDIM, kDIM, 0);
    gemm_bias_act<<<dim3(kBS / 64, kG / 64),   blk, 0, stream>>>(x, Wg1, bg1, G1, kBS, kG,   kDIM, 1);
    // Head-softmax gate
    gate_kernel<<<kBS / 16, 32, 0, stream>>>(G1, Wg2, bg2, gate);
    // Gated flash attention
    attention_kernel<<<dim3(kS / 128, kH, kB), blk, 0, stream>>>(Qb, Kbuf, Vbuf, gate, attn);
    // Output projection
    gemm_bias_act<<<dim3(kBS / 64, kDIM / 64), blk, 0, stream>>>(attn, Wo, bo, out, kBS, kDIM, kDIM, 0);
}